// DBN_60894046322756
// MI455X (gfx1250) — compile-verified
//
#include <hip/hip_runtime.h>

typedef __attribute__((ext_vector_type(2))) float v2f;
typedef __attribute__((ext_vector_type(8))) float v8f;
typedef int i4 __attribute__((vector_size(16)));   // matches builtin param type

#define G      32
#define HW     50176                 // 224*224
#define NCOL   (32 * HW)             // columns per group = 1,605,632
#define TILE1  128                   // covariance tile width (columns)
#define TILES1 (NCOL / TILE1)        // 12544
#define PAD1   132                   // 16B-aligned rows, conflict-free strided col reads
#define TILE3  64                    // apply tile width
#define TILES3 (NCOL / TILE3)        // 25088
#define PAD3   68
#define EPS    1e-5f
#define NS_ITER 16

// ws layout (floats): S[1024] | gsum[32] | wm[1024] | mean[32] | bias[32]
#define WS_S    0
#define WS_SUM  1024
#define WS_WM   1056
#define WS_MEAN 2080
#define WS_BIAS 2112
#define WS_TOT  2144

#if __has_builtin(__builtin_amdgcn_global_load_async_to_lds_b128) && \
    __has_builtin(__builtin_amdgcn_s_wait_asynccnt)
#define HAVE_ASYNC 1
#else
#define HAVE_ASYNC 0
#endif

#define GLOBAL_AS __attribute__((address_space(1)))
#define LDS_AS    __attribute__((address_space(3)))

__device__ __forceinline__ void async_copy16(const float* gp, float* lp) {
#if HAVE_ASYNC
    __builtin_amdgcn_global_load_async_to_lds_b128(
        (GLOBAL_AS i4*)gp, (LDS_AS i4*)lp, 0, 0);
#else
    *(float4*)lp = *(const float4*)gp;
#endif
}

__device__ __forceinline__ void async_wait_all() {
#if HAVE_ASYNC
    __builtin_amdgcn_s_wait_asynccnt(0);
#endif
}

__global__ void dbn_zero(float* w, int n) {
    int i = blockIdx.x * blockDim.x + threadIdx.x;
    if (i < n) w[i] = 0.0f;
}

// ---------------------------------------------------------------------------
// Pass 1: Gram matrix S = sum_j x_j x_j^T (32x32) and per-group sums.
// Block = 128 threads (4 waves). Tiles of 32x128 columns staged HBM->LDS via
// async b128 copies; each wave accumulates one 16x16 quadrant with
// V_WMMA_F32_16X16X4_F32; per-group sums re-read from LDS as b128.
// ---------------------------------------------------------------------------
__global__ __launch_bounds__(128) void dbn_cov(const float* __restrict__ x,
                                               float* __restrict__ S,
                                               float* __restrict__ gsum) {
    __shared__ float tile[G][PAD1];
    __shared__ float sred[G];

    const int tid  = threadIdx.x;          // 0..127
    const int lane = tid & 31;
    const int wave = tid >> 5;             // 0..3
    const int g0   = (wave >> 1) * 16;     // quadrant row base (A rows)
    const int g1   = (wave & 1) * 16;      // quadrant col base (B cols)
    const int m    = lane & 15;
    const int kk   = lane >> 4;            // lane half -> K pair / row half
    const int srow = tid >> 2;             // sum row 0..31 (4 threads per row)
    const int scol = (tid & 3) * 32;       // 32-col sub-chunk

    float sumloc = 0.0f;
    v8f acc = {0.f, 0.f, 0.f, 0.f, 0.f, 0.f, 0.f, 0.f};

    for (int t = blockIdx.x; t < TILES1; t += gridDim.x) {
        const int q  = t / (HW / TILE1);
        const int r0 = (t % (HW / TILE1)) * TILE1;
        const size_t base = (size_t)q * G * HW + (size_t)r0;

        __syncthreads();                   // prior readers done before overwrite
#pragma unroll
        for (int j = 0; j < 8; ++j) {      // 1024 float4 chunks / 128 threads
            int cid = j * 128 + tid;
            int g   = cid >> 5;
            int c4  = (cid & 31) * 4;
            async_copy16(x + base + (size_t)g * HW + c4, &tile[g][c4]);
        }
        async_wait_all();
        __syncthreads();

        // per-group sums from LDS (b128 reads)
#pragma unroll
        for (int j = 0; j < 8; ++j) {
            float4 v = *(const float4*)&tile[srow][scol + j * 4];
            sumloc += (v.x + v.y) + (v.z + v.w);
        }

#pragma unroll
        for (int jc = 0; jc < TILE1; jc += 4) {
            v2f a, b;
            // A (16x4): lane = 16*(K/2)+M, vgpr = K%2 -> K = 2*kk + v
            a.x = tile[g0 + m][jc + 2 * kk + 0];
            a.y = tile[g0 + m][jc + 2 * kk + 1];
            // B (4x16): vgpr v, row = v + 2*(lane/16)
            b.x = tile[g1 + m][jc + 2 * kk + 0];
            b.y = tile[g1 + m][jc + 2 * kk + 1];
            acc = __builtin_amdgcn_wmma_f32_16x16x4_f32(
                false, a, false, b, (short)0, acc, false, false);
        }
    }

    // fold per-thread group sums
    if (tid < G) sred[tid] = 0.0f;
    __syncthreads();
    atomicAdd(&sred[srow], sumloc);
    __syncthreads();
    if (tid < G) atomicAdd(&gsum[tid], sred[tid]);

    // fold Gram quadrant: C/D layout: vgpr v -> row v + 8*(lane/16), col lane%16
#pragma unroll
    for (int v = 0; v < 8; ++v) {
        int row = g0 + v + 8 * kk;
        int col = g1 + m;
        atomicAdd(&S[row * G + col], acc[v]);
    }
}

// ---------------------------------------------------------------------------
// Pass 2: sigma = S/n - mu mu^T + eps I ; Newton-Schulz -> wm = sigma^{-1/2};
// also bias = wm @ mean. One block, 1024 threads, thread (r,c) per element.
// ---------------------------------------------------------------------------
__global__ __launch_bounds__(1024) void dbn_solve(const float* __restrict__ S,
                                                  const float* __restrict__ gsum,
                                                  float* __restrict__ wm,
                                                  float* __restrict__ meanOut,
                                                  float* __restrict__ biasOut) {
    __shared__ float Y[G][G + 1], Z[G][G + 1], T[G][G + 1];
    __shared__ float Y2[G][G + 1], Z2[G][G + 1];
    __shared__ float mloc[G];
    __shared__ float trS;

    const int r = threadIdx.x >> 5;
    const int c = threadIdx.x & 31;
    const float n = (float)NCOL;

    const float mr = gsum[r] / n;
    const float mc = gsum[c] / n;
    float sig = S[r * G + c] / n - mr * mc + ((r == c) ? EPS : 0.0f);
    if (c == 0) { meanOut[r] = mr; mloc[r] = mr; }

    Y[r][c] = sig;
    Z[r][c] = (r == c) ? 1.0f : 0.0f;
    __syncthreads();

    if (threadIdx.x == 0) {
        float tr = 0.0f;
        for (int i = 0; i < G; ++i) tr += Y[i][i];
        trS = tr;
    }
    __syncthreads();
    const float tr = trS;
    Y[r][c] = sig / tr;                 // A = sigma/tr : eigenvalues in (0,1]
    __syncthreads();

    for (int it = 0; it < NS_ITER; ++it) {
        float a = 0.0f;
#pragma unroll
        for (int k = 0; k < G; ++k) a += Z[r][k] * Y[k][c];
        T[r][c] = ((r == c) ? 3.0f : 0.0f) - a;
        __syncthreads();
        float ay = 0.0f, az = 0.0f;
#pragma unroll
        for (int k = 0; k < G; ++k) {
            ay += Y[r][k] * T[k][c];
            az += T[r][k] * Z[k][c];
        }
        Y2[r][c] = 0.5f * ay;
        Z2[r][c] = 0.5f * az;
        __syncthreads();
        Y[r][c] = Y2[r][c];
        Z[r][c] = Z2[r][c];
        __syncthreads();
    }

    // wm = Z * rsqrt(tr);   bias = wm @ mean
    const float wv = Z[r][c] * rsqrtf(tr);
    wm[r * G + c] = wv;
    Y2[r][c] = wv;
    __syncthreads();
    if (c == 0) {
        float b = 0.0f;
#pragma unroll
        for (int k = 0; k < G; ++k) b += Y2[r][k] * mloc[k];
        biasOut[r] = b;
    }
}

// ---------------------------------------------------------------------------
// Pass 3: y = wm @ x - bias (== wm @ (x - mean)). Block = 256 threads
// (8 waves); raw 32x64 x-tiles staged async HBM->LDS; each wave computes one
// 16x16 output subtile via 8 WMMAs over K=32; bias folded at store.
// ---------------------------------------------------------------------------
__global__ __launch_bounds__(256) void dbn_apply(const float* __restrict__ x,
                                                 const float* __restrict__ wm,
                                                 const float* __restrict__ biasv,
                                                 float* __restrict__ out) {
    __shared__ float xt[G][PAD3];
    __shared__ float wmS[G][G + 1];

    const int tid  = threadIdx.x;          // 0..255
    const int lane = tid & 31;
    const int wave = tid >> 5;             // 0..7
    const int m0   = (wave >> 2) * 16;     // output-row half
    const int c0   = (wave & 3) * 16;      // column sub-slot
    const int m    = lane & 15;
    const int kk   = lane >> 4;

    for (int i = tid; i < G * G; i += 256) wmS[i >> 5][i & 31] = wm[i];

    // per-lane bias for the 8 D rows this lane owns (row = m0 + v + 8*kk)
    float biasr[8];
#pragma unroll
    for (int v = 0; v < 8; ++v) biasr[v] = biasv[m0 + v + 8 * kk];
    __syncthreads();

    for (int t = blockIdx.x; t < TILES3; t += gridDim.x) {
        const int q  = t / (HW / TILE3);
        const int r0 = (t % (HW / TILE3)) * TILE3;
        const size_t base = (size_t)q * G * HW + (size_t)r0;

        __syncthreads();
#pragma unroll
        for (int j = 0; j < 2; ++j) {      // 512 float4 chunks / 256 threads
            int cid = j * 256 + tid;
            int g   = cid >> 4;
            int c4  = (cid & 15) * 4;
            async_copy16(x + base + (size_t)g * HW + c4, &xt[g][c4]);
        }
        async_wait_all();
        __syncthreads();

        v8f acc = {0.f, 0.f, 0.f, 0.f, 0.f, 0.f, 0.f, 0.f};
#pragma unroll
        for (int k0 = 0; k0 < G; k0 += 4) {
            v2f a, b;
            a.x = wmS[m0 + m][k0 + 2 * kk + 0];
            a.y = wmS[m0 + m][k0 + 2 * kk + 1];
            b.x = xt[k0 + 2 * kk + 0][c0 + m];
            b.y = xt[k0 + 2 * kk + 1][c0 + m];
            acc = __builtin_amdgcn_wmma_f32_16x16x4_f32(
                false, a, false, b, (short)0, acc, false, false);
        }

        // D layout: vgpr v -> row m0 + v + 8*kk, col c0 + m (64B coalesced runs)
#pragma unroll
        for (int v = 0; v < 8; ++v) {
            int row = m0 + v + 8 * kk;
            int col = c0 + m;
            out[base + (size_t)row * HW + col] = acc[v] - biasr[v];
        }
    }
}

extern "C" void kernel_launch(void* const* d_in, const int* in_sizes, int n_in,
                              void* d_out, int out_size, void* d_ws, size_t ws_size,
                              hipStream_t stream) {
    (void)in_sizes; (void)n_in; (void)out_size; (void)ws_size;
    const float* x = (const float*)d_in[0];
    float* out = (float*)d_out;
    float* w   = (float*)d_ws;

    float* S     = w + WS_S;
    float* gsum  = w + WS_SUM;
    float* wmbuf = w + WS_WM;
    float* meanv = w + WS_MEAN;
    float* biasv = w + WS_BIAS;

    dbn_zero<<<(WS_TOT + 255) / 256, 256, 0, stream>>>(w, WS_TOT);
    dbn_cov<<<1568, 128, 0, stream>>>(x, S, gsum);
    dbn_solve<<<1, 1024, 0, stream>>>(S, gsum, wmbuf, meanv, biasv);
    dbn_apply<<<3136, 256, 0, stream>>>(x, wmbuf, biasv, out);
}